// MultiHeadAttention_17119739642013
// MI455X (gfx1250) — compile-verified
//
#include <hip/hip_runtime.h>

typedef __attribute__((ext_vector_type(2))) float v2f;
typedef __attribute__((ext_vector_type(4))) float v4f;
typedef __attribute__((ext_vector_type(8))) float v8f;

#define S_LEN  2048
#define DMODEL 1024
#define NH     16
#define HD     64
#define LDSP   68          // padded LDS row stride (floats): bank-conflict-free
#define NEG_BIG (-1.0e30f)

// --- CDNA5 async global->LDS path (probe via __has_builtin) -----------------
#if defined(__AMDGCN__) && __has_builtin(__builtin_amdgcn_global_load_async_to_lds_b128)
#define USE_ASYNC 1
#else
#define USE_ASYNC 0
#endif

#if USE_ASYNC
// pointee type leaked by clang diagnostic: int __vector_size__(16)
typedef __attribute__((__vector_size__(4 * sizeof(int)))) int i4vec;
typedef __attribute__((address_space(1))) i4vec* gas_i4p;   // global int4*
typedef __attribute__((address_space(3))) i4vec* las_i4p;   // lds int4*
#if __has_builtin(__builtin_amdgcn_s_wait_asynccnt)
#define WAIT_ASYNC(n) __builtin_amdgcn_s_wait_asynccnt(n)
#else
#define WAIT_ASYNC(n) asm volatile("s_wait_asynccnt %0" ::"i"(n))
#endif
#else
#define WAIT_ASYNC(n) do { } while (0)
#endif

// copy 16 bytes global -> LDS (async DMA when available)
__device__ __forceinline__ void cp16(const float* g, float* l) {
#if USE_ASYNC
    __builtin_amdgcn_global_load_async_to_lds_b128((gas_i4p)g, (las_i4p)l, 0, 0);
#else
    *(v4f*)l = *(const v4f*)g;
#endif
}

// ---------------------------------------------------------------------------
// Kernel 1: flash attention, fp32 WMMA (V_WMMA_F32_16X16X4_F32), double-
// buffered K/V tiles staged with async global->LDS DMA.
// grid = (S/64, H, B), block = 128 (4 waves).
// ---------------------------------------------------------------------------
__global__ __launch_bounds__(128) void mha_flash_f32(
    const float* __restrict__ Q, const float* __restrict__ K,
    const float* __restrict__ V, float* __restrict__ ctx)
{
    __shared__ float s_kv[4 * 64 * LDSP];   // [buf0:K,V][buf1:K,V]
    __shared__ float s_p[64 * LDSP];        // 4 waves x 16 rows (P relayout)

    const int tid  = threadIdx.x;
    const int lane = tid & 31;
    const int wave = tid >> 5;
    const int half = lane >> 4;
    const int l16  = lane & 15;

    const int qt = blockIdx.x;
    const int h  = blockIdx.y;
    const int b  = blockIdx.z;

    const int    qrow0   = qt * 64 + wave * 16;
    const size_t base_bh = ((size_t)b * S_LEN) * DMODEL + (size_t)h * HD;

    float* const kb0 = s_kv;
    float* const vb0 = s_kv + 1 * 64 * LDSP;
    float* const kb1 = s_kv + 2 * 64 * LDSP;
    float* const vb1 = s_kv + 3 * 64 * LDSP;

    const int r  = tid >> 1;            // staging row
    const int c0 = (tid & 1) * 32;      // staging col base

    auto stage = [&](float* kd, float* vd, int kt) {
        const float* kp = K + base_bh + (size_t)(kt * 64 + r) * DMODEL + c0;
        const float* vp = V + base_bh + (size_t)(kt * 64 + r) * DMODEL + c0;
        float* sk = kd + r * LDSP + c0;
        float* sv = vd + r * LDSP + c0;
        #pragma unroll
        for (int i = 0; i < 8; ++i) {
            cp16(kp + 4 * i, sk + 4 * i);
            cp16(vp + 4 * i, sv + 4 * i);
        }
    };

    // --- Q in WMMA A-layout registers, pre-scaled by 1/sqrt(64) ---
    v2f qa[16];
    {
        const float* qp = Q + base_bh + (size_t)(qrow0 + l16) * DMODEL;
        #pragma unroll
        for (int c = 0; c < 16; ++c) {
            v2f t = *(const v2f*)(qp + 4 * c + 2 * half);
            qa[c].x = t.x * 0.125f;
            qa[c].y = t.y * 0.125f;
        }
    }

    v8f o[4];
    #pragma unroll
    for (int d = 0; d < 4; ++d) o[d] = (v8f){0.f,0.f,0.f,0.f,0.f,0.f,0.f,0.f};
    float mrow[8], lrow[8];
    #pragma unroll
    for (int i = 0; i < 8; ++i) { mrow[i] = NEG_BIG; lrow[i] = 0.0f; }

    float* pw = s_p + wave * 16 * LDSP;

    stage(kb0, vb0, 0);   // prologue: tile 0 in flight

    for (int kt = 0; kt < S_LEN / 64; ++kt) {
        const int cur = kt & 1;
        float* sk = cur ? kb1 : kb0;
        float* sv = cur ? vb1 : vb0;
        if (kt + 1 < S_LEN / 64) {
            stage(cur ? kb0 : kb1, cur ? vb0 : vb1, kt + 1);  // overlap next tile
            WAIT_ASYNC(16);   // 16 newest in flight -> current tile landed
        } else {
            WAIT_ASYNC(0);
        }
        __syncthreads();      // current tile visible to whole workgroup

        // --- S = (Q/sqrt(hd)) * K^T : 4 col-tiles x 16 K-chunks of WMMA ---
        v8f sc[4];
        #pragma unroll
        for (int nt = 0; nt < 4; ++nt) {
            v8f acc = (v8f){0.f,0.f,0.f,0.f,0.f,0.f,0.f,0.f};
            const float* kbp = sk + (nt * 16 + l16) * LDSP + 2 * half;
            #pragma unroll
            for (int c = 0; c < 16; ++c) {
                v2f bv = *(const v2f*)(kbp + 4 * c);
                acc = __builtin_amdgcn_wmma_f32_16x16x4_f32(
                    false, qa[c], false, bv, (short)0, acc, false, false);
            }
            sc[nt] = acc;
        }

        // --- online softmax (rows replicated across each 16-lane half) ---
        float nm[8];
        #pragma unroll
        for (int i = 0; i < 8; ++i) {
            float m = fmaxf(fmaxf(sc[0][i], sc[1][i]), fmaxf(sc[2][i], sc[3][i]));
            m = fmaxf(m, __shfl_xor(m, 1));
            m = fmaxf(m, __shfl_xor(m, 2));
            m = fmaxf(m, __shfl_xor(m, 4));
            m = fmaxf(m, __shfl_xor(m, 8));
            nm[i] = fmaxf(mrow[i], m);
        }
        float rs[8];
        #pragma unroll
        for (int i = 0; i < 8; ++i) rs[i] = 0.0f;
        #pragma unroll
        for (int nt = 0; nt < 4; ++nt) {
            #pragma unroll
            for (int i = 0; i < 8; ++i) {
                float p = __expf(sc[nt][i] - nm[i]);
                rs[i] += p;
                pw[(i + 8 * half) * LDSP + nt * 16 + l16] = p;  // C->A relayout
            }
        }
        #pragma unroll
        for (int i = 0; i < 8; ++i) {
            float rsum = rs[i];
            rsum += __shfl_xor(rsum, 1);
            rsum += __shfl_xor(rsum, 2);
            rsum += __shfl_xor(rsum, 4);
            rsum += __shfl_xor(rsum, 8);
            float alpha = __expf(mrow[i] - nm[i]);
            lrow[i] = lrow[i] * alpha + rsum;
            mrow[i] = nm[i];
            #pragma unroll
            for (int dt = 0; dt < 4; ++dt) o[dt][i] *= alpha;
        }

        // --- O += P * V (same-wave LDS is in-order; no block barrier) ---
        #pragma unroll
        for (int c = 0; c < 16; ++c) {
            v2f av = *(const v2f*)(pw + l16 * LDSP + 4 * c + 2 * half);
            #pragma unroll
            for (int dt = 0; dt < 4; ++dt) {
                v2f bv;
                bv.x = sv[(4 * c + 2 * half)     * LDSP + dt * 16 + l16];
                bv.y = sv[(4 * c + 2 * half + 1) * LDSP + dt * 16 + l16];
                o[dt] = __builtin_amdgcn_wmma_f32_16x16x4_f32(
                    false, av, false, bv, (short)0, o[dt], false, false);
            }
        }
        __syncthreads();  // all waves done with buffer `cur` before reuse
    }

    // --- normalize and write ctx in (B,S,D) layout ---
    #pragma unroll
    for (int i = 0; i < 8; ++i) {
        float inv = 1.0f / lrow[i];
        float* op = ctx + base_bh + (size_t)(qrow0 + i + 8 * half) * DMODEL;
        #pragma unroll
        for (int dt = 0; dt < 4; ++dt)
            op[dt * 16 + l16] = o[dt][i] * inv;
    }
}

// ---------------------------------------------------------------------------
// Kernel 2: out = ctx(4096x1024) @ W^T + b, fp32 WMMA, 64x64 block tiles,
// double-buffered async LDS staging.
// grid = (D/64, B*S/64), block = 128 (4 waves).
// ---------------------------------------------------------------------------
__global__ __launch_bounds__(128) void mha_proj_f32(
    const float* __restrict__ ctx, const float* __restrict__ W,
    const float* __restrict__ bias, float* __restrict__ out)
{
    __shared__ float s_ab[4 * 64 * LDSP];   // [buf0:A,W][buf1:A,W]

    const int tid  = threadIdx.x;
    const int lane = tid & 31;
    const int wave = tid >> 5;
    const int half = lane >> 4;
    const int l16  = lane & 15;

    const int ntile = blockIdx.x;
    const int mtile = blockIdx.y;

    float* const ab0 = s_ab;
    float* const wb0 = s_ab + 1 * 64 * LDSP;
    float* const ab1 = s_ab + 2 * 64 * LDSP;
    float* const wb1 = s_ab + 3 * 64 * LDSP;

    const int r  = tid >> 1;
    const int c0 = (tid & 1) * 32;

    auto stage = [&](float* ad, float* wd, int kc) {
        const float* ap = ctx + (size_t)(mtile * 64 + r) * DMODEL + kc * 64 + c0;
        const float* wp = W   + (size_t)(ntile * 64 + r) * DMODEL + kc * 64 + c0;
        float* sa = ad + r * LDSP + c0;
        float* sw = wd + r * LDSP + c0;
        #pragma unroll
        for (int i = 0; i < 8; ++i) {
            cp16(ap + 4 * i, sa + 4 * i);
            cp16(wp + 4 * i, sw + 4 * i);
        }
    };

    v8f acc[4];
    #pragma unroll
    for (int nt = 0; nt < 4; ++nt) acc[nt] = (v8f){0.f,0.f,0.f,0.f,0.f,0.f,0.f,0.f};

    stage(ab0, wb0, 0);

    for (int kc = 0; kc < DMODEL / 64; ++kc) {
        const int cur = kc & 1;
        float* sa = cur ? ab1 : ab0;
        float* sw = cur ? wb1 : wb0;
        if (kc + 1 < DMODEL / 64) {
            stage(cur ? ab0 : ab1, cur ? wb0 : wb1, kc + 1);
            WAIT_ASYNC(16);
        } else {
            WAIT_ASYNC(0);
        }
        __syncthreads();

        const float* arow = sa + (wave * 16 + l16) * LDSP + 2 * half;
        #pragma unroll
        for (int c = 0; c < 16; ++c) {
            v2f av = *(const v2f*)(arow + 4 * c);
            #pragma unroll
            for (int nt = 0; nt < 4; ++nt) {
                // B[k][n] = W[n][k] -> contiguous along k in s_w row n
                v2f bv = *(const v2f*)(sw + (nt * 16 + l16) * LDSP + 4 * c + 2 * half);
                acc[nt] = __builtin_amdgcn_wmma_f32_16x16x4_f32(
                    false, av, false, bv, (short)0, acc[nt], false, false);
            }
        }
        __syncthreads();
    }

    #pragma unroll
    for (int nt = 0; nt < 4; ++nt) {
        const int coln = ntile * 64 + nt * 16 + l16;
        const float bv = bias[coln];
        #pragma unroll
        for (int i = 0; i < 8; ++i) {
            const int row = mtile * 64 + wave * 16 + i + 8 * half;
            out[(size_t)row * DMODEL + coln] = acc[nt][i] + bv;
        }
    }
}

// ---------------------------------------------------------------------------
extern "C" void kernel_launch(void* const* d_in, const int* in_sizes, int n_in,
                              void* d_out, int out_size, void* d_ws, size_t ws_size,
                              hipStream_t stream)
{
    const float* Q    = (const float*)d_in[0];
    const float* K    = (const float*)d_in[1];
    const float* V    = (const float*)d_in[2];
    const float* W    = (const float*)d_in[3];
    const float* bias = (const float*)d_in[4];
    float* out = (float*)d_out;
    float* ctx = (float*)d_ws;   // needs 2*2048*1024*4 = 16 MB scratch

    dim3 g1(S_LEN / 64, NH, 2);
    mha_flash_f32<<<g1, dim3(128), 0, stream>>>(Q, K, V, ctx);

    dim3 g2(DMODEL / 64, (2 * S_LEN) / 64);
    mha_proj_f32<<<g2, dim3(128), 0, stream>>>(ctx, W, bias, out);
}